// ChamferDistance_88837103551002
// MI455X (gfx1250) — compile-verified
//
#include <hip/hip_runtime.h>

typedef __attribute__((ext_vector_type(2))) float v2f;
typedef __attribute__((ext_vector_type(8))) float v8f;

#define CH 2048            // database points staged per chunk (32 KB LDS)
#define WPB 8              // waves (16-query tiles) per block

// Chamfer nearest-neighbor distances, both directions in one launch
// (blockIdx.z selects direction).  For each query q, min over all database
// points p of |q-p|^2.  One wave32 owns a 16-query tile.  WMMA computes
// |p|^2 - 2 q.p for a 16x16 tile per instruction:
//     A row = (-2qx,-2qy,-2qz,1),  B col = (px,py,pz,|p|^2)
// The B-matrix is staged in LDS pre-swizzled into the exact per-lane WMMA
// register layout (lanes 0-15 -> (x,y) = K0/K1, lanes 16-31 -> (z,|p|^2) =
// K2/K3), so the hot loop is ds_load_2addr_b64 + 2x v_wmma + 8x v_min3.
// |q|^2 is added once in the epilogue (it cannot change the argmin).
__global__ __launch_bounds__(256) void chamfer_nn_wmma(
    const float* __restrict__ xyz1,   // [B, n, 3]
    const float* __restrict__ xyz2,   // [B, m, 3]
    float* __restrict__ out,          // [B*n] dist1 ++ [B*m] dist2
    int n, int m)
{
    // [tile][lane] -> B-matrix register pair for that lane.
    __shared__ float2 smem[(CH / 16) * 32];

    const int lane  = threadIdx.x & 31;
    const int wave  = threadIdx.x >> 5;
    const int batch = blockIdx.y;
    const int dir   = blockIdx.z;

    // dir 0: queries = xyz1 (n), database = xyz2 (m) -> out[0 .. B*n)
    // dir 1: queries = xyz2 (m), database = xyz1 (n) -> out[B*n .. B*n+B*m)
    const float* qpts = dir ? xyz2 : xyz1;
    const float* dpts = dir ? xyz1 : xyz2;
    const int nq = dir ? m : n;
    const int md = dir ? n : m;
    float* outp = out + (dir ? (size_t)gridDim.y * n : 0);

    const int qbase = (blockIdx.x * WPB + wave) * 16;
    const int col   = lane & 15;
    const bool lo   = lane < 16;

    // Per-lane query point: lane l holds query (qbase + l%16).
    const int qi = qbase + col;
    float qx = 0.f, qy = 0.f, qz = 0.f;
    if (qi < nq) {
        const float* qp = qpts + ((size_t)batch * nq + qi) * 3;
        qx = qp[0]; qy = qp[1]; qz = qp[2];
    }
    const float a2 = qx * qx + qy * qy + qz * qz;

    // A-matrix (16x4 f32): VGPR0 = K0 (lanes 0-15) / K2 (lanes 16-31), VGPR1 = K1 / K3.
    v2f amat;
    amat.x = lo ? (-2.0f * qx) : (-2.0f * qz);
    amat.y = lo ? (-2.0f * qy) : 1.0f;

    float rmin[8];
    #pragma unroll
    for (int v = 0; v < 8; ++v) rmin[v] = 3.4e38f;

    const float* dbase = dpts + (size_t)batch * md * 3;

    for (int c0 = 0; c0 < md; c0 += CH) {
        // Cooperative stage, pre-swizzled: point j of tile t lands in
        // smem[t*32 + j] = (x,y) and smem[t*32 + j + 16] = (z,|p|^2).
        for (int i = threadIdx.x; i < CH; i += blockDim.x) {
            const int p = c0 + i;
            float x = 0.f, y = 0.f, z = 0.f, b2 = 3.4e38f;  // pad never wins min
            if (p < md) {
                const float* bp = dbase + (size_t)p * 3;
                x = bp[0]; y = bp[1]; z = bp[2];
                b2 = x * x + y * y + z * z;
            }
            const int t = i >> 4, j = i & 15;
            smem[t * 32 + j]      = make_float2(x, y);
            smem[t * 32 + j + 16] = make_float2(z, b2);
        }
        __syncthreads();

        #pragma unroll 8
        for (int t = 0; t < CH / 16; ++t) {
            // One ds_load_b64 per lane (compiler pairs them into
            // ds_load_2addr_b64); 32 lanes x 8 B span all 64 banks.
            const float2 bp = smem[t * 32 + lane];
            v2f bmat;
            bmat.x = bp.x;
            bmat.y = bp.y;
            v8f acc = {};
            // D = A x B + 0  ->  D[row][col] = |p_col|^2 - 2 * q_row . p_col
            acc = __builtin_amdgcn_wmma_f32_16x16x4_f32(
                false, amat, false, bmat, (short)0, acc, false, false);
            #pragma unroll
            for (int v = 0; v < 8; ++v) rmin[v] = fminf(rmin[v], acc[v]);
        }
        __syncthreads();
    }

    // Epilogue: add |q|^2 (row-dependent), min-reduce across each 16-lane half,
    // lanes 0 / 16 store rows v / v+8.
    #pragma unroll
    for (int v = 0; v < 8; ++v) {
        const float a2lo = __shfl(a2, v, 32);       // |q_v|^2     (held by lane v)
        const float a2hi = __shfl(a2, v + 8, 32);   // |q_{v+8}|^2 (held by lane v+8)
        float r = rmin[v] + (lo ? a2lo : a2hi);
        #pragma unroll
        for (int mask = 1; mask <= 8; mask <<= 1)
            r = fminf(r, __shfl_xor(r, mask, 32));
        const int row = lo ? v : (v + 8);
        if (col == 0 && (qbase + row) < nq)
            outp[(size_t)batch * nq + qbase + row] = r;
    }
}

extern "C" void kernel_launch(void* const* d_in, const int* in_sizes, int n_in,
                              void* d_out, int out_size, void* d_ws, size_t ws_size,
                              hipStream_t stream) {
    const float* xyz1 = (const float*)d_in[0];
    const float* xyz2 = (const float*)d_in[1];
    float* out = (float*)d_out;

    const int B = 4;
    const int n = in_sizes[0] / (B * 3);   // 16384
    const int m = in_sizes[1] / (B * 3);   // 16384

    // n == m for this problem, so one fused launch covers both directions.
    const int qtiles = (n + 15) / 16;
    dim3 block(256);
    dim3 grid((qtiles + WPB - 1) / WPB, B, 2);
    chamfer_nn_wmma<<<grid, block, 0, stream>>>(xyz1, xyz2, out, n, m);
}